// DiscreteSimulation_65609920413847
// MI455X (gfx1250) — compile-verified
//
#include <hip/hip_runtime.h>
#include <stdint.h>

// Fisher-KPP reaction-diffusion, persistent LDS-resident kernel for gfx1250.
// One workgroup per batch image; u[b] (256x256 f32 = 256KB) lives in LDS
// (320KB/WGP on CDNA5) for all t[b] steps. Bulk staging via the Tensor Data
// Mover (tensor_load_to_lds / tensor_store_from_lds, TENSORcnt), params
// warmed into L2 with global_prefetch_b8.
//
// Each thread owns 64 cells as 32 adjacent pairs (float2): intra-row
// neighbors come free from the pair load; up/down are b64 LDS loads.
// Each step runs as two half-image phases (rows 0..127 / 128..255) with the
// old row 127 parked in a 1KB LDS side buffer, so staging needs only
// 16 float2 registers per phase (32 VGPRs) -> no spill pressure.

#define HH   256
#define WW   256
#define CELLS (HH * WW)          // 65536
#define HALF  (CELLS / 2)        // 32768 (rows 0..127)
#define TPB   1024               // 32 waves (wave32)
#define KP    16                 // pairs per thread per phase

typedef unsigned int u32;
typedef u32 u32x4 __attribute__((ext_vector_type(4)));
typedef int i32x4 __attribute__((ext_vector_type(4)));
typedef int i32x8 __attribute__((ext_vector_type(8)));

// Build TDM descriptor groups for a 2D 256x256 f32 tensor, tile = whole tensor.
// Bit layout per CDNA5 ISA 8.3/8.4:
//  G0: [1:0]=count=1, [63:32]=lds_addr, [120:64]=global_addr, [127:126]=type=2
//  G1: [17:16]=data_size(2 -> 4B), [79:48]=tensor_dim0, [111:80]=tensor_dim1,
//      [127:112]=tile_dim0, [143:128]=tile_dim1, [207:160]=tensor_dim0_stride
__device__ __forceinline__ void tdm_desc_2d(u32x4& g0, i32x8& g1,
                                            unsigned long long gaddr,
                                            u32 lds_byte_off) {
  g0.x = 1u;                                   // count=1, is_restore=0
  g0.y = lds_byte_off;                         // lds_addr
  g0.z = (u32)(gaddr & 0xffffffffu);           // global_addr[31:0]
  g0.w = (u32)((gaddr >> 32) & 0x01ffffffu)    // global_addr[56:32]
       | (2u << 30);                           // type = 2 ("image")
  g1[0] = (int)(2u << 16);                     // workgroup_mask=0, data_size=4B
  g1[1] = (int)(256u << 16);                   // tensor_dim0[15:0] (bits 63:48)
  g1[2] = (int)(256u << 16);                   // dim0[31:16]=0 | tensor_dim1[15:0]
  g1[3] = (int)(256u << 16);                   // dim1[31:16]=0 | tile_dim0=256
  g1[4] = (int)256;                            // tile_dim1=256, tile_dim2=0
  g1[5] = (int)256;                            // tensor_dim0_stride[31:0] = 256
  g1[6] = 0;                                   // stride0[47:32]=0, stride1[15:0]=0
  g1[7] = 0;
}

__device__ __forceinline__ float2 pair_update(float2 cc, float l, float r,
                                              float2 up, float2 dn,
                                              float2 D2, float2 R2) {
  // zero-padded SAME conv: absent neighbors contribute 0, center -4 stays
  const float lap0 = (l + cc.y) + (up.x + dn.x) - 4.0f * cc.x;
  const float lap1 = (cc.x + r) + (up.y + dn.y) - 4.0f * cc.y;
  float v0 = cc.x + D2.x * lap0 + R2.x * cc.x * (1.0f - cc.x);
  float v1 = cc.y + D2.y * lap1 + R2.y * cc.y * (1.0f - cc.y);
  v0 = fminf(fmaxf(v0, 0.0f), 1.0f);
  v1 = fminf(fmaxf(v1, 0.0f), 1.0f);
  return make_float2(v0, v1);
}

extern "C" __global__ __launch_bounds__(TPB, 1)
void fkpp_persistent(const float* __restrict__ u,
                     const float* __restrict__ params,
                     const int* __restrict__ t,
                     float* __restrict__ out) {
  __shared__ float su[CELLS];                  // 256 KB, whole image resident
  __shared__ float sbound[WW];                 // old row 127 (phase handoff)

  const int tid = threadIdx.x;
  const int b   = blockIdx.x;

  const float* __restrict__ ub = u + (size_t)b * CELLS;
  const float* __restrict__ Db = params + (size_t)b * 2 * CELLS;  // D plane
  const float* __restrict__ Rb = Db + CELLS;                      // rho plane
  float* __restrict__ ob       = out + (size_t)b * CELLS;

  const int steps = t[b];                      // per-batch truncation: clip is
                                               // idempotent for masked steps

  // Warm both param planes (512KB) into L2 (-> global_prefetch_b8).
#pragma unroll
  for (int k = 0; k < 4; ++k)
    __builtin_prefetch(Db + 32u * (u32)(tid + k * TPB), 0, 3);

  // ---- Stage u[b] global -> LDS via the Tensor Data Mover ----------------
#if __has_builtin(__builtin_amdgcn_tensor_load_to_lds)
  if (tid < 32) {                              // one wave issues the TDM op
    u32x4 g0; i32x8 g1;
    tdm_desc_2d(g0, g1, (unsigned long long)(uintptr_t)ub, 0u);
    i32x4 z4 = (i32x4){0, 0, 0, 0};            // groups 2/3 unused (2D tensor)
    i32x8 z8 = (i32x8){0, 0, 0, 0, 0, 0, 0, 0};
    __builtin_amdgcn_tensor_load_to_lds(g0, g1, z4, z4, z8, 0);
    __builtin_amdgcn_s_wait_tensorcnt(0);
  }
#else
  for (int k = tid; k < CELLS / 4; k += TPB)
    ((float4*)su)[k] = ((const float4*)ub)[k];
#endif
  __syncthreads();

  // ---- Time stepping -----------------------------------------------------
  const int  p0    = 2 * tid;                  // pair base offset
  const int  col0  = p0 & (WW - 1);            // loop-invariant column
  const bool has_l = (col0 != 0);
  const bool has_r = (col0 != WW - 2);

  float2 un[KP];                               // 32 VGPR staging per phase

  for (int s = 0; s < steps; ++s) {
    // Park old row 127 so phase B can read it after phase A overwrites it.
    if (tid < WW / 2)
      ((float2*)sbound)[tid] = ((const float2*)(su + 127 * WW))[tid];

    // -- Phase A: rows 0..127 (reads old rows 0..128) --
#pragma unroll
    for (int k = 0; k < KP; ++k) {
      const int p = p0 + k * 2048;             // 8 rows per k step
      const float2 cc = *(const float2*)(su + p);
      const float  l  = has_l ? su[p - 1] : 0.0f;
      const float  r  = has_r ? su[p + 2] : 0.0f;
      float2 up;
      if (k == 0)                              // rows 0..7: row 0 pads with 0
        up = (p >= WW) ? *(const float2*)(su + p - WW) : make_float2(0.f, 0.f);
      else
        up = *(const float2*)(su + p - WW);
      const float2 dn = *(const float2*)(su + p + WW);   // row+1 <= 128 exists
      const float2 D2 = *(const float2*)(Db + p);        // L2-resident
      const float2 R2 = *(const float2*)(Rb + p);
      un[k] = pair_update(cc, l, r, up, dn, D2, R2);
    }
    __syncthreads();   // all phase-A reads (and sbound copy) complete
#pragma unroll
    for (int k = 0; k < KP; ++k)
      *(float2*)(su + p0 + k * 2048) = un[k];

    // -- Phase B: rows 128..255 (reads old rows 128..255 + sbound) --
    //    Disjoint from phase-A writes (rows <= 127), so no barrier needed
    //    between the A-writeback above and these reads.
#pragma unroll
    for (int k = 0; k < KP; ++k) {
      const int p = HALF + p0 + k * 2048;
      const float2 cc = *(const float2*)(su + p);
      const float  l  = has_l ? su[p - 1] : 0.0f;
      const float  r  = has_r ? su[p + 2] : 0.0f;
      float2 up;
      if (k == 0)                              // rows 128..135
        up = (p0 < WW) ? *(const float2*)(sbound + p0)   // row 128: old row 127
                       : *(const float2*)(su + p - WW);
      else
        up = *(const float2*)(su + p - WW);
      float2 dn;
      if (k == KP - 1)                         // rows 248..255: row 255 pads 0
        dn = (p + WW < CELLS) ? *(const float2*)(su + p + WW)
                              : make_float2(0.f, 0.f);
      else
        dn = *(const float2*)(su + p + WW);
      const float2 D2 = *(const float2*)(Db + p);
      const float2 R2 = *(const float2*)(Rb + p);
      un[k] = pair_update(cc, l, r, up, dn, D2, R2);
    }
    __syncthreads();   // all phase-B reads complete
#pragma unroll
    for (int k = 0; k < KP; ++k)
      *(float2*)(su + HALF + p0 + k * 2048) = un[k];
    __syncthreads();   // all writes visible before next step
  }

  // ---- Drain LDS -> out[b] via the Tensor Data Mover ---------------------
#if __has_builtin(__builtin_amdgcn_tensor_store_from_lds)
  if (tid < 32) {
    u32x4 g0; i32x8 g1;
    tdm_desc_2d(g0, g1, (unsigned long long)(uintptr_t)ob, 0u);
    i32x4 z4 = (i32x4){0, 0, 0, 0};
    i32x8 z8 = (i32x8){0, 0, 0, 0, 0, 0, 0, 0};
    __builtin_amdgcn_tensor_store_from_lds(g0, g1, z4, z4, z8, 0);
    __builtin_amdgcn_s_wait_tensorcnt(0);
  }
#else
  for (int k = tid; k < CELLS / 4; k += TPB)
    ((float4*)ob)[k] = ((const float4*)su)[k];
#endif
}

extern "C" void kernel_launch(void* const* d_in, const int* in_sizes, int n_in,
                              void* d_out, int out_size, void* d_ws,
                              size_t ws_size, hipStream_t stream) {
  (void)n_in; (void)out_size; (void)d_ws; (void)ws_size;
  const float* u      = (const float*)d_in[0];   // (B,1,256,256) f32
  const float* params = (const float*)d_in[1];   // (B,2,256,256) f32
  const int*   t      = (const int*)d_in[2];     // (B,) i32
  float*       out    = (float*)d_out;           // (B,1,256,256) f32
  const int B = in_sizes[2];                     // t has B elements
  fkpp_persistent<<<dim3(B), dim3(TPB), 0, stream>>>(u, params, t, out);
}